// M_ATTA_79602923864454
// MI455X (gfx1250) — compile-verified
//
#include <hip/hip_runtime.h>

#define N_CLASSES 1000
#define N_AUGMS   64
#define N_TILES   63              // ceil(1000/16)
#define PADDED    (N_TILES * 16)  // 1008
#define BLOCK     1024
#define NWAVES    (BLOCK / 32)
#define MAXK      128             // omega sequence length bound (underflow by ~k=101)

typedef __attribute__((ext_vector_type(2))) float v2f;
typedef __attribute__((ext_vector_type(8))) float v8f;

__device__ __forceinline__ float waveMax(float v) {
#pragma unroll
  for (int off = 16; off > 0; off >>= 1) v = fmaxf(v, __shfl_xor(v, off, 32));
  return v;
}

__device__ __forceinline__ float waveSum(float v) {
#pragma unroll
  for (int off = 16; off > 0; off >>= 1) v += __shfl_xor(v, off, 32);
  return v;
}

__device__ __forceinline__ void waveArgmax(float& v, int& i) {
#pragma unroll
  for (int off = 16; off > 0; off >>= 1) {
    float ov = __shfl_xor(v, off, 32);
    int   oi = __shfl_xor(i, off, 32);
    if (ov > v || (ov == v && oi < i)) { v = ov; i = oi; }
  }
}

__device__ __forceinline__ float blockMax(float v, float* sv) {
  const int lane = threadIdx.x & 31, wv = threadIdx.x >> 5;
  v = waveMax(v);
  if (lane == 0) sv[wv] = v;
  __syncthreads();
  if (wv == 0) {
    float t = (lane < NWAVES) ? sv[lane] : -__builtin_inff();
    t = waveMax(t);
    if (lane == 0) sv[0] = t;
  }
  __syncthreads();
  float r = sv[0];
  __syncthreads();
  return r;
}

__device__ __forceinline__ float blockSum(float v, float* sv) {
  const int lane = threadIdx.x & 31, wv = threadIdx.x >> 5;
  v = waveSum(v);
  if (lane == 0) sv[wv] = v;
  __syncthreads();
  if (wv == 0) {
    float t = (lane < NWAVES) ? sv[lane] : 0.0f;
    t = waveSum(t);
    if (lane == 0) sv[0] = t;
  }
  __syncthreads();
  float r = sv[0];
  __syncthreads();
  return r;
}

__device__ __forceinline__ int blockArgmax(float v, int i, float* sv, int* si) {
  const int lane = threadIdx.x & 31, wv = threadIdx.x >> 5;
  waveArgmax(v, i);
  if (lane == 0) { sv[wv] = v; si[wv] = i; }
  __syncthreads();
  if (wv == 0) {
    float tv = (lane < NWAVES) ? sv[lane] : -__builtin_inff();
    int   ti = (lane < NWAVES) ? si[lane] : 0x7fffffff;
    waveArgmax(tv, ti);
    if (lane == 0) si[0] = ti;
  }
  __syncthreads();
  int r = si[0];
  __syncthreads();
  return r;
}

extern "C" __global__ __launch_bounds__(BLOCK, 1)
void atta_fused(const float* __restrict__ p0, const float* __restrict__ P,
                const float* __restrict__ W, const float* __restrict__ omega,
                float* __restrict__ out) {
  __shared__ float pp[PADDED];    // row sums of W*P
  __shared__ float q[PADDED];     // softmax(pp) == WP_nom
  __shared__ float p0s[PADDED];   // cached p0
  __shared__ float omg[MAXK];     // exact f32 omega sequence
  __shared__ int   matchk[MAXK];  // match(omega_k)
  __shared__ float sv[NWAVES];
  __shared__ int   si[NWAVES];
  __shared__ float s_om0;
  __shared__ int   s_kuf;         // first k where (omega_k - 0.01f) < 0.001f
  __shared__ float s_omf;

  const int tid  = threadIdx.x;
  const int lane = tid & 31;
  const int wv   = tid >> 5;

  if (tid < PADDED) p0s[tid] = (tid < N_CLASSES) ? p0[tid] : 0.0f;
  if (tid == 0) s_om0 = fminf(fmaxf(omega[0], 0.0f), 1.0f);  // clip(omega, 0, 1)

  // ---- Phase 1: pp[i] = sum_j W[i][j]*P[i][j] via V_WMMA_F32_16X16X4_F32 ----
  // A fragment (16x4 f32): lanes 0-15 -> M=lane, K={0,1}; lanes 16-31 -> M=lane-16, K={2,3}.
  // Each A element is a VALU pre-reduction of 16 products; B = all-ones so D = rowsum(A).
  v2f bfrag;
  bfrag.x = 1.0f;
  bfrag.y = 1.0f;
  for (int tile = wv; tile < N_TILES; tile += NWAVES) {
    const int row = tile * 16 + (lane & 15);
    const int kb  = (lane >> 4) << 1;  // 0 or 2
    float s0 = 0.0f, s1 = 0.0f;
    if (row < N_CLASSES) {
      const float* Wr = W + row * N_AUGMS;
      const float* Pr = P + row * N_AUGMS;
#pragma unroll
      for (int j = 0; j < 16; ++j) {
        s0 = fmaf(Wr[kb * 16 + j],       Pr[kb * 16 + j],       s0);
        s1 = fmaf(Wr[(kb + 1) * 16 + j], Pr[(kb + 1) * 16 + j], s1);
      }
    }
    v2f afrag;
    afrag.x = s0;
    afrag.y = s1;
    v8f c = {};
    c = __builtin_amdgcn_wmma_f32_16x16x4_f32(false, afrag, false, bfrag,
                                              (short)0, c, false, false);
    // D layout (16x16 f32): VGPR r -> M=r (lanes 0-15) / M=8+r (lanes 16-31).
    if (lane == 0) {
#pragma unroll
      for (int r = 0; r < 8; ++r) pp[tile * 16 + r] = c[r];
    } else if (lane == 16) {
#pragma unroll
      for (int r = 0; r < 8; ++r) pp[tile * 16 + 8 + r] = c[r];
    }
  }
  __syncthreads();

  // ---- Phase 2: q = softmax(pp) over classes ----
  float myv = (tid < N_CLASSES) ? pp[tid] : -__builtin_inff();
  float m = blockMax(myv, sv);
  float e = (tid < N_CLASSES) ? expf(myv - m) : 0.0f;
  float s = blockSum(e, sv);
  if (tid < PADDED) q[tid] = (tid < N_CLASSES) ? (e / s) : 0.0f;

  // ---- Phase 3: p0c = argmax(p0), first-index tie-break ----
  float v0 = (tid < N_CLASSES) ? p0s[tid] : -__builtin_inff();
  const int p0c = blockArgmax(v0, tid, sv, si);  // uniform across block
  __syncthreads();                               // q[] visible to all

  // ---- Phase 4a: exact f32 omega sequence (independent of match results) ----
  if (tid == 0) {
    float o  = s_om0;
    int  kuf = MAXK - 1;
    for (int k = 0; k < MAXK; ++k) {
      omg[k] = o;
      const float dec = o - 0.01f;     // exact reference recurrence in f32
      if (dec < 0.001f) { kuf = k; break; }
      o = dec;
    }
    s_kuf = kuf;
  }
  __syncthreads();
  const int kuf = s_kuf;

  // ---- Phase 4b: evaluate match(omega_k) for all k in parallel ----
  // match <=> argmax((1-w)p0 + w q) == p0c
  //       <=> !exists i: pv_i > pv_p0c  ||  (pv_i == pv_p0c && i < p0c)
  // Same f32 blend expression as the reference => bit-exact comparisons.
  const float p0pc = p0s[p0c];
  const float qpc  = q[p0c];
  for (int k = wv; k < MAXK; k += NWAVES) {
    if (k <= kuf) {
      const float omk = omg[k];
      const float one_m = 1.0f - omk;
      const float mpc = one_m * p0pc + omk * qpc;
      int bad = 0;
      for (int i = lane; i < N_CLASSES; i += 32) {
        const float pv = one_m * p0s[i] + omk * q[i];
        bad |= (pv > mpc) || (pv == mpc && i < p0c);
      }
      const int anybad = __any(bad);   // wave32 vote, no LDS round-trips
      if (lane == 0) matchk[k] = !anybad;
    }
  }
  __syncthreads();

  // ---- Phase 4c: k* = first k with match, else underflow k ----
  if (tid == 0) {
    int kstar = kuf;
    for (int k = 0; k <= kuf; ++k) {
      if (matchk[k]) { kstar = k; break; }
    }
    s_omf = omg[kstar];
  }
  __syncthreads();

  // ---- Output: p_new at the terminating iteration, shape (1, 1000) ----
  const float omf = s_omf;
  if (tid < N_CLASSES) out[tid] = (1.0f - omf) * p0s[tid] + omf * q[tid];
}

extern "C" void kernel_launch(void* const* d_in, const int* in_sizes, int n_in,
                              void* d_out, int out_size, void* d_ws, size_t ws_size,
                              hipStream_t stream) {
  const float* p0 = (const float*)d_in[0];   // (1000, 1)
  const float* P  = (const float*)d_in[1];   // (1000, 64)
  const float* W  = (const float*)d_in[2];   // (1000, 64)
  const float* om = (const float*)d_in[3];   // (1,)
  float* out = (float*)d_out;                // (1, 1000)
  (void)in_sizes; (void)n_in; (void)out_size; (void)d_ws; (void)ws_size;
  atta_fused<<<1, BLOCK, 0, stream>>>(p0, P, W, om, out);
}